// LearnableSDFGrid_16621523435902
// MI455X (gfx1250) — compile-verified
//
#include <hip/hip_runtime.h>
#include <hip/hip_bf16.h>

// LearnableSDFGrid ray-march on MI455X (gfx1250).
//
// Shape: R=32768 rays x S=128 steps. Outputs (f32, concatenated flat):
//   sdf       [R,S,8]    = 33,554,432 floats
//   positions [R,S,8,3]  = 100,663,296 floats
//   valid     [R,S]      = 4,194,304 floats (1.0/0.0)
// ~528 MiB of streaming stores vs a 64 MiB L2-resident SDF grid -> pure
// store-bandwidth problem (~24us floor at 23.3 TB/s). Strategy:
//   - NT stores (keep SDF grid hot in the 192MB L2, stream outputs past it)
//   - b128 vector stores, one thread per (ray,step), fully coalesced
//   - per-ray ray data staged via gfx1250 global_load_async_to_lds_b32
//   - 8-corner gather via 8 scalar b32 loads, skipped for invalid waves

#define NRAYS 32768
#define NSTEP 128

typedef float v4f __attribute__((ext_vector_type(4)));

__global__ __launch_bounds__(NSTEP) void sdf_march_kernel(
    const float* __restrict__ origins,   // [R,3]
    const float* __restrict__ dirs,      // [R,3]
    const float* __restrict__ grid,      // [256,256,256]
    float* __restrict__ out_sdf,         // [R,S,8]
    float* __restrict__ out_pos,         // [R,S,8,3]
    float* __restrict__ out_valid)       // [R,S]
{
    __shared__ float ray[8];             // o.xyz, d.xyz (+pad)
    const int r = blockIdx.x;
    const int s = threadIdx.x;

#if defined(__AMDGCN__) && \
    __has_builtin(__builtin_amdgcn_global_load_async_to_lds_b32) && \
    __has_builtin(__builtin_amdgcn_s_wait_asynccnt)
    // gfx1250 async path: one DMA-style copy of the 6 per-ray floats into LDS,
    // broadcast to all 128 step-threads (ASYNCcnt-tracked).
    if (s < 6) {
        const float* src = (s < 3) ? (origins + 3 * r + s) : (dirs + 3 * r + (s - 3));
        __builtin_amdgcn_global_load_async_to_lds_b32(
            (__attribute__((address_space(1))) int*)src,
            (__attribute__((address_space(3))) int*)&ray[s],
            0, 0);
    }
    __builtin_amdgcn_s_wait_asynccnt(0);
    __syncthreads();
#else
    if (s < 6) ray[s] = (s < 3) ? origins[3 * r + s] : dirs[3 * r + (s - 3)];
    __syncthreads();
#endif

    const float ox = ray[0], oy = ray[1], oz = ray[2];
    const float ux = ray[3], uy = ray[4], uz = ray[5];

    // ray_box_intersection, faithful including the tmin-reuse-in-tmax quirk.
    const float ivx = 1.0f / ux, ivy = 1.0f / uy, ivz = 1.0f / uz;
    const float ax = (-1.0f - ox) * ivx, bx = (1.0f - ox) * ivx;
    const float ay = (-1.0f - oy) * ivy, by = (1.0f - oy) * ivy;
    const float az = (-1.0f - oz) * ivz, bz = (1.0f - oz) * ivz;
    const float t0 = fmaxf(fmaxf(fminf(ax, bx), fminf(ay, by)), fminf(az, bz));
    const float t1 = fminf(fminf(fmaxf(t0, bx), fmaxf(t0, by)), fmaxf(t0, bz));
    const bool  hit = (t1 >= t0);

    const float t = t0 + 0.02f * (float)s;
    const bool  smask = (t < t1) && hit;

    // query_voxels: truncating index like .astype(int32)
    const float vox = 2.0f / 255.0f;     // (max-min)/(n-1)
    const float px = ox + t * ux, py = oy + t * uy, pz = oz + t * uz;
    const int ix = (int)((px + 1.0f) / vox);
    const int iy = (int)((py + 1.0f) / vox);
    const int iz = (int)((pz + 1.0f) / vox);
    const bool inb = ((unsigned)ix <= 254u) && ((unsigned)iy <= 254u) && ((unsigned)iz <= 254u);
    const bool valid = inb && smask;

    const size_t base = (size_t)r * NSTEP + (size_t)s;
    float* sp = out_sdf + base * 8;      // 32B aligned
    float* pp = out_pos + base * 24;     // 96B -> 16B aligned

    if (valid) {
        // in-bounds => clip(ix,0,254) == ix
        const float* g = grid + (((size_t)ix << 16) + ((unsigned)iy << 8) + (unsigned)iz);
        const float s000 = g[0],             s001 = g[1];           // (dx,dy)=(0,0), dz=0/1
        const float s010 = g[256],           s011 = g[257];         // dy=1
        const float s100 = g[65536],         s101 = g[65537];       // dx=1
        const float s110 = g[65536 + 256],   s111 = g[65536 + 257]; // dx=dy=1

        // corner order c: (dx,dy,dz) = (c&1, (c>>1)&1, (c>>2)&1)
        v4f sv0 = { s000, s100, s010, s110 };   // c0..c3 (dz=0)
        v4f sv1 = { s001, s101, s011, s111 };   // c4..c7 (dz=1)
        __builtin_nontemporal_store(sv0, (v4f*)sp);
        __builtin_nontemporal_store(sv1, (v4f*)(sp + 4));

        const float x0 = -1.0f + (float)ix * vox, x1 = -1.0f + (float)(ix + 1) * vox;
        const float y0 = -1.0f + (float)iy * vox, y1 = -1.0f + (float)(iy + 1) * vox;
        const float z0 = -1.0f + (float)iz * vox, z1 = -1.0f + (float)(iz + 1) * vox;
        // flat [8,3]: c0(x0,y0,z0) c1(x1,y0,z0) c2(x0,y1,z0) c3(x1,y1,z0)
        //             c4(x0,y0,z1) c5(x1,y0,z1) c6(x0,y1,z1) c7(x1,y1,z1)
        v4f p0 = { x0, y0, z0, x1 };
        v4f p1 = { y0, z0, x0, y1 };
        v4f p2 = { z0, x1, y1, z0 };
        v4f p3 = { x0, y0, z1, x1 };
        v4f p4 = { y0, z1, x0, y1 };
        v4f p5 = { z1, x1, y1, z1 };
        __builtin_nontemporal_store(p0, (v4f*)(pp +  0));
        __builtin_nontemporal_store(p1, (v4f*)(pp +  4));
        __builtin_nontemporal_store(p2, (v4f*)(pp +  8));
        __builtin_nontemporal_store(p3, (v4f*)(pp + 12));
        __builtin_nontemporal_store(p4, (v4f*)(pp + 16));
        __builtin_nontemporal_store(p5, (v4f*)(pp + 20));
        __builtin_nontemporal_store(1.0f, out_valid + base);
    } else {
        const v4f zero = { 0.0f, 0.0f, 0.0f, 0.0f };
        __builtin_nontemporal_store(zero, (v4f*)sp);
        __builtin_nontemporal_store(zero, (v4f*)(sp + 4));
        __builtin_nontemporal_store(zero, (v4f*)(pp +  0));
        __builtin_nontemporal_store(zero, (v4f*)(pp +  4));
        __builtin_nontemporal_store(zero, (v4f*)(pp +  8));
        __builtin_nontemporal_store(zero, (v4f*)(pp + 12));
        __builtin_nontemporal_store(zero, (v4f*)(pp + 16));
        __builtin_nontemporal_store(zero, (v4f*)(pp + 20));
        __builtin_nontemporal_store(0.0f, out_valid + base);
    }
}

extern "C" void kernel_launch(void* const* d_in, const int* in_sizes, int n_in,
                              void* d_out, int out_size, void* d_ws, size_t ws_size,
                              hipStream_t stream) {
    (void)in_sizes; (void)n_in; (void)out_size; (void)d_ws; (void)ws_size;
    const float* origins = (const float*)d_in[0];
    const float* dirs    = (const float*)d_in[1];
    const float* grid    = (const float*)d_in[2];

    float* out       = (float*)d_out;
    float* out_sdf   = out;                                      // R*S*8
    float* out_pos   = out_sdf + (size_t)NRAYS * NSTEP * 8;      // R*S*8*3
    float* out_valid = out_pos + (size_t)NRAYS * NSTEP * 24;     // R*S

    sdf_march_kernel<<<NRAYS, NSTEP, 0, stream>>>(origins, dirs, grid,
                                                  out_sdf, out_pos, out_valid);
}